// GAT_36481452212962
// MI455X (gfx1250) — compile-verified
//
#include <hip/hip_runtime.h>
#include <hip/hip_bf16.h>
#include <math.h>

typedef __attribute__((ext_vector_type(2))) float v2f;
typedef __attribute__((ext_vector_type(8))) float v8f;

#define MID 64
#define NCLS 40
#define NCLSP 48                 // NCLS padded to 3 WMMA N-tiles
#define GAT_NEG_SLOPE 0.2f

// ---- monotone float <-> uint mapping for atomicMax on floats ----
__device__ __forceinline__ unsigned f2ord(float f) {
    unsigned b = __float_as_uint(f);
    return (b & 0x80000000u) ? ~b : (b | 0x80000000u);
}
__device__ __forceinline__ float ord2f(unsigned k) {
    unsigned b = (k & 0x80000000u) ? (k ^ 0x80000000u) : ~k;
    return __uint_as_float(b);
}

// =====================================================================
// GEMM: Hout[M x 64] = act(Xin) @ W[K x 64], act = relu(x + bprev) if ACT
// One wave computes a 16(M) x 64(N) strip via 4x V_WMMA_F32_16X16X4_F32.
// Block = 256 threads = 8 wave32s -> 128 rows per block.
// W staged TRANSPOSED in LDS (sWt[col][k], stride K+4): a lane's B
// fragment (W[f0][col], W[f0+1][col]) is one aligned ds_load_b64, and
// since (K+4) mod 64 == 4 the two half-waves hit disjoint bank sets.
// =====================================================================
template<int K, bool ACT>
__global__ __launch_bounds__(256)
void gat_gemm(const float* __restrict__ Xin, const float* __restrict__ W,
              const float* __restrict__ bprev,
              float* __restrict__ Hout, int M)
{
    constexpr int STK = K + 4;
    __shared__ float sWt[MID * STK];
    __shared__ float sB[K];
    for (int i = threadIdx.x; i < K * MID; i += 256) {
        int k = i >> 6, col = i & 63;
        sWt[col * STK + k] = W[i];
    }
    if (ACT) {
        for (int i = threadIdx.x; i < K; i += 256) sB[i] = bprev[i];
    }
    __syncthreads();

    const int wave = threadIdx.x >> 5;
    const int lane = threadIdx.x & 31;
    const int r    = lane & 15;   // M row within tile (A); N col within tile (B/C/D)
    const int half = lane >> 4;   // selects K pair (A/B); selects M+8 (C/D)
    const int row0 = blockIdx.x * 128 + wave * 16;

    int arow = row0 + r;
    if (arow >= M) arow = M - 1;          // clamp; stores are guarded
    const float* xrow = Xin + (long long)arow * K;

    v8f acc[4] = {};                      // 4 N-tiles of 16 -> 64 cols

    for (int kk = 0; kk < K; kk += 4) {
        const int f0 = kk + 2 * half;     // this lane's K offset (A-matrix layout)
        v2f a = *(const v2f*)&xrow[f0];   // 8B aligned: f0 even, row 256B-aligned
        if (ACT) {
            a.x = fmaxf(a.x + sB[f0], 0.0f);
            a.y = fmaxf(a.y + sB[f0 + 1], 0.0f);
        }
        #pragma unroll
        for (int nt = 0; nt < 4; ++nt) {
            v2f b = *(const v2f*)&sWt[(nt * 16 + r) * STK + f0];  // ds_load_b64
            acc[nt] = __builtin_amdgcn_wmma_f32_16x16x4_f32(
                false, a, false, b, (short)0, acc[nt], false, false);
        }
    }

    // C/D layout: VGPR v, lanes 0-15 -> M=v ; lanes 16-31 -> M=v+8 ; N = nt*16 + r
    #pragma unroll
    for (int nt = 0; nt < 4; ++nt) {
        #pragma unroll
        for (int v = 0; v < 8; ++v) {
            int grow = row0 + v + 8 * half;
            if (grow < M)
                Hout[(long long)grow * MID + nt * 16 + r] = acc[nt][v];
        }
    }
}

// es[n] = h[n] . a_src ; ed[n] = h[n] . a_dst
__global__ __launch_bounds__(256)
void gat_proj(const float* __restrict__ H, const float* __restrict__ avs,
              const float* __restrict__ avd, float* __restrict__ es,
              float* __restrict__ ed, int N)
{
    int n = blockIdx.x * 256 + threadIdx.x;
    if (n >= N) return;
    const float* h = H + (long long)n * MID;
    float s = 0.0f, d = 0.0f;
    #pragma unroll
    for (int k = 0; k < MID; ++k) {
        float v = h[k];
        s += v * avs[k];
        d += v * avd[k];
    }
    es[n] = s;
    ed[n] = d;
}

__global__ __launch_bounds__(256)
void gat_init(float* __restrict__ agg, unsigned* __restrict__ nmaxk,
              float* __restrict__ den, int N)
{
    long long i = (long long)blockIdx.x * 256 + threadIdx.x;
    if (i < (long long)N * MID) agg[i] = 0.0f;
    if (i < N) {
        nmaxk[i] = f2ord(-3.0e38f);
        den[i] = 0.0f;
    }
}

// pass 1: e = leaky_relu(es[src]+ed[dst]); segment-max over dst via ordered atomicMax
__global__ __launch_bounds__(256)
void gat_edge_max(const long long* __restrict__ ei, int E, int N,
                  const float* __restrict__ es, const float* __restrict__ ed,
                  float* __restrict__ eval, unsigned* __restrict__ nmaxk)
{
    long long i = (long long)blockIdx.x * 256 + threadIdx.x;
    long long TOT = (long long)E + N;
    if (i >= TOT) return;
    long long s, d;
    if (i < E) { s = ei[i]; d = ei[E + i]; } else { s = d = i - E; }
    float v = es[s] + ed[d];
    v = (v > 0.0f) ? v : GAT_NEG_SLOPE * v;
    eval[i] = v;
    atomicMax(nmaxk + d, f2ord(v));
}

// pass 2: ee = exp(e - max[dst]); segment-sum
__global__ __launch_bounds__(256)
void gat_edge_expsum(const long long* __restrict__ ei, int E, int N,
                     float* __restrict__ eval, const unsigned* __restrict__ nmaxk,
                     float* __restrict__ den)
{
    long long i = (long long)blockIdx.x * 256 + threadIdx.x;
    long long TOT = (long long)E + N;
    if (i >= TOT) return;
    long long d;
    if (i < E) { d = ei[E + i]; } else { d = i - E; }
    float v  = eval[i];
    float m  = ord2f(nmaxk[d]);
    float ee = expf(v - m);
    eval[i]  = ee;
    atomicAdd(den + d, ee);
}

// pass 3: agg[dst] += alpha * h[src]; 16 lanes per edge, float4 each (coalesced row)
__global__ __launch_bounds__(256)
void gat_edge_agg(const long long* __restrict__ ei, int E, int N,
                  const float* __restrict__ eval, const float* __restrict__ den,
                  const float* __restrict__ H, float* __restrict__ agg)
{
    long long idx = (long long)blockIdx.x * 256 + threadIdx.x;
    long long TOT = ((long long)E + N) * 16;
    if (idx >= TOT) return;
    long long e = idx >> 4;
    int c = (int)(idx & 15);
    long long s, d;
    if (e < E) { s = ei[e]; d = ei[E + e]; } else { s = d = e - E; }
    float alpha = eval[e] / (den[d] + 1e-16f);
    const float4 hv = *(const float4*)(H + s * MID + c * 4);
    float* o = agg + d * MID + c * 4;
    atomicAdd(o + 0, alpha * hv.x);
    atomicAdd(o + 1, alpha * hv.y);
    atomicAdd(o + 2, alpha * hv.z);
    atomicAdd(o + 3, alpha * hv.w);
}

// =====================================================================
// Classifier via WMMA: out[n][c] = relu(agg[n]+b1) @ Wc + bc, 40 cols
// padded to 48 (3 N-tiles of 16). Wc staged transposed + zero-padded in
// LDS (stride 68 -> conflict-free b64). Stores guarded to col < 40.
// =====================================================================
__global__ __launch_bounds__(256)
void gat_cls(const float* __restrict__ agg, const float* __restrict__ b1,
             const float* __restrict__ Wc, const float* __restrict__ bc,
             float* __restrict__ out, int N)
{
    constexpr int STK = MID + 4;          // 68
    __shared__ float sWcT[NCLSP * STK];
    __shared__ float sbc[NCLSP];
    __shared__ float sb1[MID];
    for (int i = threadIdx.x; i < NCLSP * STK; i += 256) sWcT[i] = 0.0f;
    if (threadIdx.x < NCLSP)
        sbc[threadIdx.x] = (threadIdx.x < NCLS) ? bc[threadIdx.x] : 0.0f;
    if (threadIdx.x < MID) sb1[threadIdx.x] = b1[threadIdx.x];
    __syncthreads();
    for (int i = threadIdx.x; i < MID * NCLS; i += 256) {
        int k = i / NCLS, c = i % NCLS;
        sWcT[c * STK + k] = Wc[i];
    }
    __syncthreads();

    const int wave = threadIdx.x >> 5;
    const int lane = threadIdx.x & 31;
    const int r    = lane & 15;
    const int half = lane >> 4;
    const int row0 = blockIdx.x * 128 + wave * 16;

    int arow = row0 + r;
    if (arow >= N) arow = N - 1;
    const float* xrow = agg + (long long)arow * MID;

    v8f acc[3] = {};
    for (int kk = 0; kk < MID; kk += 4) {
        const int f0 = kk + 2 * half;
        v2f a = *(const v2f*)&xrow[f0];
        a.x = fmaxf(a.x + sb1[f0],     0.0f);
        a.y = fmaxf(a.y + sb1[f0 + 1], 0.0f);
        #pragma unroll
        for (int nt = 0; nt < 3; ++nt) {
            v2f b = *(const v2f*)&sWcT[(nt * 16 + r) * STK + f0];
            acc[nt] = __builtin_amdgcn_wmma_f32_16x16x4_f32(
                false, a, false, b, (short)0, acc[nt], false, false);
        }
    }

    #pragma unroll
    for (int nt = 0; nt < 3; ++nt) {
        const int col = nt * 16 + r;
        #pragma unroll
        for (int v = 0; v < 8; ++v) {
            int grow = row0 + v + 8 * half;
            if (grow < N && col < NCLS)
                out[(long long)grow * NCLS + col] = acc[nt][v] + sbc[col];
        }
    }
}

// in-place row-wise log_softmax over 40 classes
__global__ __launch_bounds__(256)
void gat_lsm(float* __restrict__ out, int N)
{
    int n = blockIdx.x * 256 + threadIdx.x;
    if (n >= N) return;
    float* row = out + (long long)n * NCLS;
    float m = -3.0e38f;
    #pragma unroll
    for (int c = 0; c < NCLS; ++c) m = fmaxf(m, row[c]);
    float s = 0.0f;
    #pragma unroll
    for (int c = 0; c < NCLS; ++c) s += expf(row[c] - m);
    float lse = m + logf(s);
    #pragma unroll
    for (int c = 0; c < NCLS; ++c) row[c] = row[c] - lse;
}

extern "C" void kernel_launch(void* const* d_in, const int* in_sizes, int n_in,
                              void* d_out, int out_size, void* d_ws, size_t ws_size,
                              hipStream_t stream)
{
    const float*     x   = (const float*)d_in[0];
    const long long* ei  = (const long long*)d_in[1];   // int64 [2, E]
    const float*     W0  = (const float*)d_in[2];
    const float*     as0 = (const float*)d_in[3];
    const float*     ad0 = (const float*)d_in[4];
    const float*     b0  = (const float*)d_in[5];
    const float*     W1  = (const float*)d_in[6];
    const float*     as1 = (const float*)d_in[7];
    const float*     ad1 = (const float*)d_in[8];
    const float*     b1  = (const float*)d_in[9];
    const float*     Wc  = (const float*)d_in[10];
    const float*     bc  = (const float*)d_in[11];

    const int N = in_sizes[0] / 128;
    const int E = in_sizes[1] / 2;
    const long long TOT = (long long)E + N;

    // workspace layout (floats)
    float* ws = (float*)d_ws;
    float*    H     = ws;                ws += (size_t)N * MID;   // transformed features
    float*    AGG   = ws;                ws += (size_t)N * MID;   // aggregation target
    float*    ES    = ws;                ws += N;
    float*    ED    = ws;                ws += N;
    unsigned* NMAXK = (unsigned*)ws;     ws += N;
    float*    DEN   = ws;                ws += N;
    float*    EVAL  = ws;                ws += TOT;

    const dim3 blk(256);
    const int gGemm = (N + 127) / 128;
    const int gN    = (N + 255) / 256;
    const int gInit = (int)(((long long)N * MID + 255) / 256);
    const int gE    = (int)((TOT + 255) / 256);
    const int gAgg  = (int)((TOT * 16 + 255) / 256);

    // ---- layer 1: h = x @ W0 ----
    gat_gemm<128, false><<<gGemm, blk, 0, stream>>>(x, W0, nullptr, H, N);
    gat_proj<<<gN, blk, 0, stream>>>(H, as0, ad0, ES, ED, N);
    gat_init<<<gInit, blk, 0, stream>>>(AGG, NMAXK, DEN, N);
    gat_edge_max<<<gE, blk, 0, stream>>>(ei, E, N, ES, ED, EVAL, NMAXK);
    gat_edge_expsum<<<gE, blk, 0, stream>>>(ei, E, N, EVAL, NMAXK, DEN);
    gat_edge_agg<<<gAgg, blk, 0, stream>>>(ei, E, N, EVAL, DEN, H, AGG);

    // ---- layer 2: h = relu(agg + b0) @ W1 (act fused into GEMM load) ----
    gat_gemm<64, true><<<gGemm, blk, 0, stream>>>(AGG, W1, b0, H, N);
    gat_proj<<<gN, blk, 0, stream>>>(H, as1, ad1, ES, ED, N);
    gat_init<<<gInit, blk, 0, stream>>>(AGG, NMAXK, DEN, N);
    gat_edge_max<<<gE, blk, 0, stream>>>(ei, E, N, ES, ED, EVAL, NMAXK);
    gat_edge_expsum<<<gE, blk, 0, stream>>>(ei, E, N, EVAL, NMAXK, DEN);
    gat_edge_agg<<<gAgg, blk, 0, stream>>>(ei, E, N, EVAL, DEN, H, AGG);

    // ---- classifier (WMMA, relu(agg+b1) fused) + log_softmax ----
    gat_cls<<<gGemm, blk, 0, stream>>>(AGG, b1, Wc, bc, (float*)d_out, N);
    gat_lsm<<<gN, blk, 0, stream>>>((float*)d_out, N);
}